// HOAF_85985245265995
// MI455X (gfx1250) — compile-verified
//
#include <hip/hip_runtime.h>

// out[:,c] = inp[:,c] * group_sum(c) / 32 over (32,256,56,56) fp32,
// 32 groups x 8 channels, HW plane = 56*56 = 3136.
//
// Memory-bound: 205 MB min traffic @ 23.3 TB/s ~ 8.8 us. All global traffic is
// b128 (256B per half-wave). The 8-channel group reduction rides the WMMA pipe:
// D = ones(16x4) x B(4x16) + C  via V_WMMA_F32_16X16X4_F32 gives
// D[m,n] = sum_k B[k,n] — the group sum broadcast so lane n%16 holds S for the
// spatial position it owns. Stores are non-temporal so the (L2-resident, 103MB)
// input isn't evicted by streamed output.

typedef __attribute__((ext_vector_type(2))) float v2f;
typedef __attribute__((ext_vector_type(4))) float v4f;
typedef __attribute__((ext_vector_type(8))) float v8f;

#define PLANE 3136          // 56*56
#define TILES_PER_BG 49     // 3136 / 64 spatial positions per wave-tile
#define NUM_BG 1024         // 32 batches * 32 groups
#define WAVES_PER_BLOCK 8
#define TOTAL_WAVES (NUM_BG * TILES_PER_BG)          // 50176
#define NUM_BLOCKS (TOTAL_WAVES / WAVES_PER_BLOCK)   // 6272 (exact)

__global__ __launch_bounds__(256) void
HOAF_85985245265995_kernel(const float* __restrict__ x, float* __restrict__ out) {
    const int tid  = threadIdx.x;
    const int lane = tid & 31;
    const int wave = tid >> 5;

    const int W    = blockIdx.x * WAVES_PER_BLOCK + wave;  // < TOTAL_WAVES exactly
    const int bg   = W / TILES_PER_BG;                     // (batch, group)
    const int tile = W - bg * TILES_PER_BG;

    const int half = lane >> 4;   // 0: lanes 0-15, 1: lanes 16-31
    const int ln   = lane & 15;

    // Lane owns spatial positions p..p+3 (float4-aligned: PLANE % 4 == 0)
    const int p    = tile * 64 + 4 * ln;
    const int base = bg * (8 * PLANE) + p;   // element index at channel 0

    // Half-wave channel split: half0 -> {0,1,4,5}, half1 -> {2,3,6,7}.
    // Lane pair (L, L+16) thus covers all 8 group channels at the same spatial
    // positions, which is exactly how a WMMA B-column is assembled.
    const int c01 = 2 * half;       // channels c01, c01+1
    const int c45 = 2 * half + 4;   // channels c45, c45+1

    const v4f A0 = *(const v4f*)(x + base + (c01 + 0) * PLANE);
    const v4f A1 = *(const v4f*)(x + base + (c01 + 1) * PLANE);
    const v4f A2 = *(const v4f*)(x + base + (c45 + 0) * PLANE);
    const v4f A3 = *(const v4f*)(x + base + (c45 + 1) * PLANE);

    v2f a;
    a.x = 1.0f;
    a.y = 1.0f;  // A = ones(16x4): D[m,n] = sum_k B[k,n]

    v4f O0, O1, O2, O3;
#pragma unroll
    for (int j = 0; j < 4; ++j) {
        // B operand for component j: column n sums channels {0..3} (then {4..7})
        // at spatial tile*64 + 4n + j.
        v2f b0; b0.x = A0[j]; b0.y = A1[j];
        v2f b1; b1.x = A2[j]; b1.y = A3[j];

        v8f c = {};
        c = __builtin_amdgcn_wmma_f32_16x16x4_f32(false, a, false, b0,
                                                  (short)0, c, false, false);
        c = __builtin_amdgcn_wmma_f32_16x16x4_f32(false, a, false, b1,
                                                  (short)0, c, false, false);

        // c[0]: N = lane%16 for every accumulator VGPR -> S at this lane's
        // spatial position p + j. Scale by 1/num_groups.
        const float s = c[0] * (1.0f / 32.0f);

        O0[j] = A0[j] * s;
        O1[j] = A1[j] * s;
        O2[j] = A2[j] * s;
        O3[j] = A3[j] * s;
    }

    __builtin_nontemporal_store(O0, (v4f*)(out + base + (c01 + 0) * PLANE));
    __builtin_nontemporal_store(O1, (v4f*)(out + base + (c01 + 1) * PLANE));
    __builtin_nontemporal_store(O2, (v4f*)(out + base + (c45 + 0) * PLANE));
    __builtin_nontemporal_store(O3, (v4f*)(out + base + (c45 + 1) * PLANE));
}

extern "C" void kernel_launch(void* const* d_in, const int* in_sizes, int n_in,
                              void* d_out, int out_size, void* d_ws, size_t ws_size,
                              hipStream_t stream) {
    (void)in_sizes; (void)n_in; (void)out_size; (void)d_ws; (void)ws_size;
    const float* x = (const float*)d_in[0];
    float* out = (float*)d_out;
    HOAF_85985245265995_kernel<<<NUM_BLOCKS, 256, 0, stream>>>(x, out);
}